// ModelNew_82248623719031
// MI455X (gfx1250) — compile-verified
//
#include <hip/hip_runtime.h>
#include <cstdint>
#include <cstddef>

#define BATCH   4
#define DIM     1536
#define DSTATE  16
#define SEQLEN  4096
#define CHUNKS  16
#define CLEN    (SEQLEN / CHUNKS)   /* 256 */
#define TPB     256
#define LOG2E   1.44269504088896340736f

typedef __attribute__((ext_vector_type(4))) float v4f;   // native vector for NT builtins

// ---------- CDNA5 async global -> LDS copy (ASYNCcnt-tracked) ----------
__device__ __forceinline__ void async_ld_b32_to_lds(uint32_t lds_addr, const float* gaddr) {
    // gfx1250: GLOBAL_LOAD_ASYNC_TO_LDS_B32 (LDS dest addr in VGPR, 64-bit global addr in VGPR pair)
    asm volatile("global_load_async_to_lds_b32 %0, %1, off"
                 :: "v"(lds_addr), "v"(gaddr) : "memory");
}
__device__ __forceinline__ void wait_async_zero() {
    asm volatile("s_wait_asynccnt 0" ::: "memory");
}
__device__ __forceinline__ uint32_t lds_off(const void* p) {
    // low 32 bits of the generic (flat) address == LDS byte offset on gfx1250
    return (uint32_t)(uintptr_t)p;
}

__device__ __forceinline__ float softplus_f(float z) {
    float e = __expf(z);
    float r = __logf(1.0f + e);
    return (z > 20.0f) ? z : r;
}
__device__ __forceinline__ float exp2_f(float x) {
    return __builtin_amdgcn_exp2f(x);   // v_exp_f32
}

__device__ __forceinline__ void load_a2(const float* __restrict__ A, int d, float* a2) {
    const float4* Ar = reinterpret_cast<const float4*>(A + (size_t)d * DSTATE);
#pragma unroll
    for (int q = 0; q < 4; ++q) {
        float4 v = Ar[q];
        a2[4*q+0] = v.x * LOG2E; a2[4*q+1] = v.y * LOG2E;
        a2[4*q+2] = v.z * LOG2E; a2[4*q+3] = v.w * LOG2E;
    }
}

// =====================================================================
// Pass 1: per-chunk local scan from zero state.
// Produces x_loc[b,c,d,0:16] (chunk-final state) and S[b,c,d] = sum softplus.
// =====================================================================
__global__ __launch_bounds__(TPB) void ssm_chunk_state(
    const float* __restrict__ u, const float* __restrict__ delta,
    const float* __restrict__ Bm, const float* __restrict__ A,
    const float* __restrict__ delta_bias,
    float* __restrict__ ws_x, float* __restrict__ ws_S)
{
    __shared__ float Bt[CLEN * DSTATE];   // [t][n], 16 KB

    const int tid = threadIdx.x;
    const int b   = blockIdx.z;
    const int c   = blockIdx.y;
    const int d   = blockIdx.x * TPB + tid;
    const int t0  = c * CLEN;

    // Async transposed tile: global B[b][n][t0+tid] -> LDS Bt[tid][n]
    const float* Bg = Bm + ((size_t)b * DSTATE) * SEQLEN + t0;
#pragma unroll
    for (int n = 0; n < DSTATE; ++n)
        async_ld_b32_to_lds(lds_off(&Bt[tid * DSTATE + n]), Bg + (size_t)n * SEQLEN + tid);

    float a2[DSTATE];
    load_a2(A, d, a2);
    const float  db = delta_bias[d];
    const size_t ub = ((size_t)b * DIM + d) * SEQLEN + t0;

    wait_async_zero();
    __syncthreads();

    float x[DSTATE];
#pragma unroll
    for (int n = 0; n < DSTATE; ++n) x[n] = 0.0f;
    float S = 0.0f;

    // software-pipelined: 8 timesteps per iteration, next batch prefetched
    float4 cu[2], cd[2];
    cu[0] = *reinterpret_cast<const float4*>(u + ub);
    cu[1] = *reinterpret_cast<const float4*>(u + ub + 4);
    cd[0] = *reinterpret_cast<const float4*>(delta + ub);
    cd[1] = *reinterpret_cast<const float4*>(delta + ub + 4);

    for (int tt = 0; tt < CLEN; tt += 8) {
        float4 nu[2] = {}, nd[2] = {};
        const int tn = tt + 8;
        if (tn < CLEN) {
            nu[0] = *reinterpret_cast<const float4*>(u + ub + tn);
            nu[1] = *reinterpret_cast<const float4*>(u + ub + tn + 4);
            nd[0] = *reinterpret_cast<const float4*>(delta + ub + tn);
            nd[1] = *reinterpret_cast<const float4*>(delta + ub + tn + 4);
        }
#pragma unroll
        for (int h = 0; h < 2; ++h) {
#pragma unroll
            for (int j = 0; j < 4; ++j) {
                const int   t   = tt + 4*h + j;
                const float uj  = (&cu[h].x)[j];
                const float dlt = softplus_f((&cd[h].x)[j] + db);
                S += dlt;
                const float du = dlt * uj;
                float Br[DSTATE];
                const float4* Brow = reinterpret_cast<const float4*>(&Bt[t * DSTATE]);
#pragma unroll
                for (int q = 0; q < 4; ++q)
                    *reinterpret_cast<float4*>(&Br[4*q]) = Brow[q];   // ds_load_b128 broadcast
#pragma unroll
                for (int n = 0; n < DSTATE; ++n)
                    x[n] = x[n] * exp2_f(a2[n] * dlt) + du * Br[n];
            }
        }
        cu[0] = nu[0]; cu[1] = nu[1]; cd[0] = nd[0]; cd[1] = nd[1];
    }

    float4* xo = reinterpret_cast<float4*>(ws_x + (((size_t)b * CHUNKS + c) * DIM + d) * DSTATE);
#pragma unroll
    for (int q = 0; q < 4; ++q)
        xo[q] = make_float4(x[4*q], x[4*q+1], x[4*q+2], x[4*q+3]);
    ws_S[((size_t)b * CHUNKS + c) * DIM + d] = S;
}

// =====================================================================
// Pass 2 (tiny): stitch chunk states; chunk-total decay = exp2(a2[n]*S_c).
// Writes x_init = state at chunk START.
// =====================================================================
__global__ __launch_bounds__(TPB) void ssm_chunk_combine(
    const float* __restrict__ A, const float* __restrict__ ws_x,
    const float* __restrict__ ws_S, float* __restrict__ ws_xi)
{
    const int idx = blockIdx.x * TPB + threadIdx.x;   // 0 .. BATCH*DIM-1
    const int b = idx / DIM;
    const int d = idx % DIM;

    float a2[DSTATE];
    load_a2(A, d, a2);

    float xin[DSTATE];
#pragma unroll
    for (int n = 0; n < DSTATE; ++n) xin[n] = 0.0f;

    for (int c = 0; c < CHUNKS; ++c) {
        const size_t off = (((size_t)b * CHUNKS + c) * DIM + d) * DSTATE;
        float4* xi = reinterpret_cast<float4*>(ws_xi + off);
#pragma unroll
        for (int q = 0; q < 4; ++q)
            xi[q] = make_float4(xin[4*q], xin[4*q+1], xin[4*q+2], xin[4*q+3]);

        const float S = ws_S[((size_t)b * CHUNKS + c) * DIM + d];
        float xl[DSTATE];
        const float4* xr = reinterpret_cast<const float4*>(ws_x + off);
#pragma unroll
        for (int q = 0; q < 4; ++q)
            *reinterpret_cast<float4*>(&xl[4*q]) = xr[q];
#pragma unroll
        for (int n = 0; n < DSTATE; ++n)
            xin[n] = xin[n] * exp2_f(a2[n] * S) + xl[n];
    }
}

// =====================================================================
// Pass 3: rerun local scan seeded with x_init, emit y.
// Last-use NT reads for u/delta; NT store for out (never re-read).
// =====================================================================
__global__ __launch_bounds__(TPB) void ssm_chunk_scan(
    const float* __restrict__ u, const float* __restrict__ delta,
    const float* __restrict__ Bm, const float* __restrict__ Cm,
    const float* __restrict__ A, const float* __restrict__ Dv,
    const float* __restrict__ delta_bias, const float* __restrict__ ws_xi,
    float* __restrict__ out)
{
    __shared__ float Bt[CLEN * DSTATE];
    __shared__ float Ct[CLEN * DSTATE];

    const int tid = threadIdx.x;
    const int b   = blockIdx.z;
    const int c   = blockIdx.y;
    const int d   = blockIdx.x * TPB + tid;
    const int t0  = c * CLEN;

    const float* Bg = Bm + ((size_t)b * DSTATE) * SEQLEN + t0;
    const float* Cg = Cm + ((size_t)b * DSTATE) * SEQLEN + t0;
#pragma unroll
    for (int n = 0; n < DSTATE; ++n)
        async_ld_b32_to_lds(lds_off(&Bt[tid * DSTATE + n]), Bg + (size_t)n * SEQLEN + tid);
#pragma unroll
    for (int n = 0; n < DSTATE; ++n)
        async_ld_b32_to_lds(lds_off(&Ct[tid * DSTATE + n]), Cg + (size_t)n * SEQLEN + tid);

    float a2[DSTATE];
    load_a2(A, d, a2);
    const float  db = delta_bias[d];
    const float  Dd = Dv[d];
    const size_t ub = ((size_t)b * DIM + d) * SEQLEN + t0;

    // carry-in state for this chunk
    float x[DSTATE];
    {
        const float4* xr = reinterpret_cast<const float4*>(
            ws_xi + (((size_t)b * CHUNKS + c) * DIM + d) * DSTATE);
#pragma unroll
        for (int q = 0; q < 4; ++q)
            *reinterpret_cast<float4*>(&x[4*q]) = xr[q];
    }

    wait_async_zero();
    __syncthreads();

    v4f cu[2], cd[2];
    cu[0] = __builtin_nontemporal_load(reinterpret_cast<const v4f*>(u + ub));
    cu[1] = __builtin_nontemporal_load(reinterpret_cast<const v4f*>(u + ub + 4));
    cd[0] = __builtin_nontemporal_load(reinterpret_cast<const v4f*>(delta + ub));
    cd[1] = __builtin_nontemporal_load(reinterpret_cast<const v4f*>(delta + ub + 4));

    for (int tt = 0; tt < CLEN; tt += 8) {
        v4f nu[2] = {(v4f)0.f, (v4f)0.f}, nd[2] = {(v4f)0.f, (v4f)0.f};
        const int tn = tt + 8;
        if (tn < CLEN) {
            nu[0] = __builtin_nontemporal_load(reinterpret_cast<const v4f*>(u + ub + tn));
            nu[1] = __builtin_nontemporal_load(reinterpret_cast<const v4f*>(u + ub + tn + 4));
            nd[0] = __builtin_nontemporal_load(reinterpret_cast<const v4f*>(delta + ub + tn));
            nd[1] = __builtin_nontemporal_load(reinterpret_cast<const v4f*>(delta + ub + tn + 4));
        }
#pragma unroll
        for (int h = 0; h < 2; ++h) {
            v4f yv;
#pragma unroll
            for (int j = 0; j < 4; ++j) {
                const int   t   = tt + 4*h + j;
                const float uj  = cu[h][j];
                const float dlt = softplus_f(cd[h][j] + db);
                const float du  = dlt * uj;

                float Br[DSTATE], Cr[DSTATE];
                const float4* Brow = reinterpret_cast<const float4*>(&Bt[t * DSTATE]);
                const float4* Crow = reinterpret_cast<const float4*>(&Ct[t * DSTATE]);
#pragma unroll
                for (int q = 0; q < 4; ++q) {
                    *reinterpret_cast<float4*>(&Br[4*q]) = Brow[q];
                    *reinterpret_cast<float4*>(&Cr[4*q]) = Crow[q];
                }

                float y = Dd * uj;
#pragma unroll
                for (int n = 0; n < DSTATE; ++n) {
                    x[n] = x[n] * exp2_f(a2[n] * dlt) + du * Br[n];
                    y += x[n] * Cr[n];
                }
                yv[j] = y;
            }
            __builtin_nontemporal_store(yv, reinterpret_cast<v4f*>(out + ub + tt + 4*h));
        }
        cu[0] = nu[0]; cu[1] = nu[1]; cd[0] = nd[0]; cd[1] = nd[1];
    }
}

// =====================================================================
extern "C" void kernel_launch(void* const* d_in, const int* in_sizes, int n_in,
                              void* d_out, int out_size, void* d_ws, size_t ws_size,
                              hipStream_t stream) {
    (void)in_sizes; (void)n_in; (void)out_size; (void)ws_size;
    const float* u     = (const float*)d_in[0];
    const float* delta = (const float*)d_in[1];
    const float* Bm    = (const float*)d_in[2];
    const float* Cm    = (const float*)d_in[3];
    const float* A     = (const float*)d_in[4];
    const float* Dv    = (const float*)d_in[5];
    const float* dbias = (const float*)d_in[6];
    float* out = (float*)d_out;

    float* ws    = (float*)d_ws;
    float* ws_x  = ws;                                               // [B][CH][DIM][16]
    float* ws_S  = ws_x + (size_t)BATCH * CHUNKS * DIM * DSTATE;     // [B][CH][DIM]
    float* ws_xi = ws_S + (size_t)BATCH * CHUNKS * DIM;              // [B][CH][DIM][16]

    dim3 grid(DIM / TPB, CHUNKS, BATCH);   // 6 x 16 x 4 = 384 blocks, 8 waves each
    ssm_chunk_state<<<grid, TPB, 0, stream>>>(u, delta, Bm, A, dbias, ws_x, ws_S);
    ssm_chunk_combine<<<(BATCH * DIM) / TPB, TPB, 0, stream>>>(A, ws_x, ws_S, ws_xi);
    ssm_chunk_scan<<<grid, TPB, 0, stream>>>(u, delta, Bm, Cm, A, Dv, dbias, ws_xi, out);
}